// DeformableTransformer_10763188044233
// MI455X (gfx1250) — compile-verified
//
#include <hip/hip_runtime.h>
#include <cstdint>
#include <cstddef>

// ---------------------------------------------------------------------------
// CDNA5 (gfx1250) deformable-transformer pipeline.
//  * fp32 GEMMs on the WMMA pipe (V_WMMA_F32_16X16X4_F32), 32x32 per wave
//  * tile staging via GLOBAL_LOAD_ASYNC_TO_LDS_B128 + s_wait_asynccnt
//  * gather/bilinear sampling: 1 wave per (b,h,q), lane = channel
// ---------------------------------------------------------------------------

typedef __attribute__((ext_vector_type(2))) float v2f;
typedef __attribute__((ext_vector_type(8))) float v8f;

#define GEMM_BLOCK 256
#define BM 64
#define BN 128
#define BK 64
#define LDA_PAD 68    // floats per A row in LDS (16B-aligned rows, bank-spread)
#define LDB_PAD 132   // floats per B row in LDS

__device__ __forceinline__ void async_ld_b128(uint32_t lds_off, const float* gp)
{
    // GLOBAL_LOAD_ASYNC_TO_LDS_B128 (GV form): VDST = LDS byte offset,
    // VADDR = 64-bit global address. Tracked by ASYNCcnt.
    asm volatile("global_load_async_to_lds_b128 %0, %1, off"
                 :: "v"(lds_off), "v"((unsigned long long)(uintptr_t)gp)
                 : "memory");
}

__device__ __forceinline__ void wait_async0()
{
    asm volatile("s_wait_asynccnt 0x0" ::: "memory");
}

// C(MxN) = A(MxK) @ B(KxN) + bias[N]   (fp32 WMMA 16x16x4, wave32)
// Requirements: K % BK == 0, N % 32 == 0.  M edge handled by clamp+guard.
// grid = (ceil(N/BN), ceil(M/BM)), block = 256 (8 waves, 2x4 wave grid).
__global__ __launch_bounds__(GEMM_BLOCK)
void wmma_gemm_f32(const float* __restrict__ A, const float* __restrict__ Bm,
                   const float* __restrict__ bias, float* __restrict__ C,
                   int M, int N, int K)
{
    __shared__ float ldsA[BM][LDA_PAD];
    __shared__ float ldsB[BK][LDB_PAD];

    const int tid  = threadIdx.x;
    const int wave = tid >> 5;
    const int lane = tid & 31;
    const int wm   = wave >> 2;              // 0..1  (m-quad)
    const int wn   = wave & 3;               // 0..3  (n-quad)
    const int m0   = blockIdx.y * BM;
    const int n0   = blockIdx.x * BN;
    const int row  = lane & 15;              // M-row of A frag / N-col of B,C frag
    const int kq   = (lane >> 4) << 1;       // lanes 0-15 -> K {0,1}; 16-31 -> {2,3}
    const int nw   = n0 + wn * 32;           // wave's first output column
    const bool act = (nw < N);               // wave-uniform (N % 32 == 0)

    v8f acc00 = {}, acc01 = {}, acc10 = {}, acc11 = {};

    for (int k0 = 0; k0 < K; k0 += BK) {
        // --- async stage A tile: BM x BK floats, 16B chunks, row clamp at M edge
        for (int c = tid; c < BM * (BK / 4); c += GEMM_BLOCK) {
            const int r  = c >> 4;           // BK/4 = 16 chunks per row
            const int c4 = (c & 15) << 2;
            int gr = m0 + r; if (gr > M - 1) gr = M - 1;
            async_ld_b128((uint32_t)(uintptr_t)&ldsA[r][c4],
                          A + (size_t)gr * K + (k0 + c4));
        }
        // --- async stage B tile: BK x BN floats (column clamp when BN > N)
        for (int c = tid; c < BK * (BN / 4); c += GEMM_BLOCK) {
            const int r  = c >> 5;           // BN/4 = 32 chunks per row
            const int c4 = (c & 31) << 2;
            int gc = n0 + c4; if (gc > N - 4) gc = N - 4;
            async_ld_b128((uint32_t)(uintptr_t)&ldsB[r][c4],
                          Bm + (size_t)(k0 + r) * N + gc);
        }
        wait_async0();
        __syncthreads();

        if (act) {
            #pragma unroll 4
            for (int kk = 0; kk < BK; kk += 4) {
                v2f a0, a1, b0, b1;
                // A 16x4 frags for the two 16-row subtiles
                a0.x = ldsA[wm * 32 + row][kk + kq];
                a0.y = ldsA[wm * 32 + row][kk + kq + 1];
                a1.x = ldsA[wm * 32 + 16 + row][kk + kq];
                a1.y = ldsA[wm * 32 + 16 + row][kk + kq + 1];
                // B 4x16 frags for the two 16-col subtiles
                b0.x = ldsB[kk + kq][wn * 32 + row];
                b0.y = ldsB[kk + kq + 1][wn * 32 + row];
                b1.x = ldsB[kk + kq][wn * 32 + 16 + row];
                b1.y = ldsB[kk + kq + 1][wn * 32 + 16 + row];

                acc00 = __builtin_amdgcn_wmma_f32_16x16x4_f32(false, a0, false, b0,
                                                              (short)0, acc00, false, false);
                acc01 = __builtin_amdgcn_wmma_f32_16x16x4_f32(false, a0, false, b1,
                                                              (short)0, acc01, false, false);
                acc10 = __builtin_amdgcn_wmma_f32_16x16x4_f32(false, a1, false, b0,
                                                              (short)0, acc10, false, false);
                acc11 = __builtin_amdgcn_wmma_f32_16x16x4_f32(false, a1, false, b1,
                                                              (short)0, acc11, false, false);
            }
        }
        __syncthreads();
    }

    if (act) {
        // C/D 16x16 f32 layout: VGPR j -> (M=j, N=lane) | (M=j+8, N=lane-16)
        const int   mhi = (lane >> 4) << 3;
        const float bn0 = bias[nw + row];
        const float bn1 = bias[nw + 16 + row];
        #pragma unroll
        for (int j = 0; j < 8; ++j) {
            const int mA = m0 + wm * 32 + mhi + j;
            const int mB = mA + 16;
            if (mA < M) {
                C[(size_t)mA * N + nw + row]      = acc00[j] + bn0;
                C[(size_t)mA * N + nw + 16 + row] = acc01[j] + bn1;
            }
            if (mB < M) {
                C[(size_t)mB * N + nw + row]      = acc10[j] + bn0;
                C[(size_t)mB * N + nw + 16 + row] = acc11[j] + bn1;
            }
        }
    }
}

// Softmax over 12 contiguous logits per (b,q,h), in place.
__global__ void softmax12_kernel(float* __restrict__ att, int total)
{
    const int i = blockIdx.x * blockDim.x + threadIdx.x;
    if (i >= total) return;
    float* p = att + (size_t)i * 12;
    float mx = p[0];
    #pragma unroll
    for (int j = 1; j < 12; ++j) mx = fmaxf(mx, p[j]);
    float e[12], s = 0.f;
    #pragma unroll
    for (int j = 0; j < 12; ++j) { e[j] = __expf(p[j] - mx); s += e[j]; }
    const float inv = 1.f / s;
    #pragma unroll
    for (int j = 0; j < 12; ++j) p[j] = e[j] * inv;
}

// Multi-scale deformable bilinear sampling.
// One wave per (b,h,q); lane = channel d (D=32). v is row-major (b*L, 256),
// head h occupies columns [h*32, h*32+32) -> each corner gather is one
// coalesced 128B global_load across the wave.
#define NQ  100
#define NK  17
#define NLV 3
#define NPT 4
#define NHD 8
#define LTOT 22050

__global__ __launch_bounds__(256)
void ms_deform_sample(const float* __restrict__ v,      // (b*L, 256)
                      const float* __restrict__ off,    // (b*lq, NH*NL*NP*4)
                      const float* __restrict__ aw,     // (b*lq, NH*NL*NP)
                      const float* __restrict__ ref_l,  // (b,100,3,17,2)
                      const float* __restrict__ ref_r,  // (b,100,3,17,2)
                      float* __restrict__ tmp,          // (b*lq, 256)
                      int LQ)
{
    const int wave = threadIdx.x >> 5;
    const int lane = threadIdx.x & 31;
    const int gid  = blockIdx.x * 8 + wave;         // (b*NH + h)*LQ + q

    const int q  = gid % LQ;
    const int bh = gid / LQ;
    const int h  = bh % NHD;
    const int b  = bh / NHD;
    const int qi = q / NK;
    const int ki = q % NK;

    const float* offp = off + ((size_t)(b * LQ + q) * NHD + h) * (NLV * NPT * 4);
    const float* awp  = aw  + ((size_t)(b * LQ + q) * NHD + h) * (NLV * NPT);

    const int HL[NLV] = {100, 50, 25};
    const int WL[NLV] = {168, 84, 42};

    float acc = 0.f;
    int lvl_start = 0;

    #pragma unroll
    for (int li = 0; li < NLV; ++li) {
        const int Hl = HL[li], Wl = WL[li];
        const float fW = (float)Wl, fH = (float)Hl;

        const size_t rbase = ((((size_t)b * NQ + qi) * NLV + li) * NK + ki) * 2;
        const float rxl = ref_l[rbase + 0], ryl = ref_l[rbase + 1];
        const float rxr = ref_r[rbase + 0], ryr = ref_r[rbase + 1];

        const float* vlev = v + ((size_t)b * LTOT + lvl_start) * 256 + h * 32 + lane;

        #pragma unroll
        for (int p = 0; p < 2 * NPT; ++p) {
            const int  pp   = p & 3;
            const bool left = (p < NPT);
            const float w   = awp[li * NPT + pp];
            const float* o  = offp + ((size_t)li * NPT + pp) * 4 + (left ? 0 : 2);

            const float lx = (left ? rxl : rxr) + o[0] / fW;
            const float ly = (left ? ryl : ryr) + o[1] / fH;
            const float x  = lx * fW - 0.5f;
            const float y  = ly * fH - 0.5f;

            const float x0f = floorf(x), y0f = floorf(y);
            const float tx = x - x0f,   ty = y - y0f;
            const int x0 = (int)x0f,    y0 = (int)y0f;

            float s = 0.f;
            #pragma unroll
            for (int cy = 0; cy < 2; ++cy) {
                #pragma unroll
                for (int cx = 0; cx < 2; ++cx) {
                    const int xi = x0 + cx, yi = y0 + cy;
                    const bool valid = (xi >= 0) & (xi < Wl) & (yi >= 0) & (yi < Hl);
                    const int xc = xi < 0 ? 0 : (xi > Wl - 1 ? Wl - 1 : xi);
                    const int yc = yi < 0 ? 0 : (yi > Hl - 1 ? Hl - 1 : yi);
                    const float g = vlev[(size_t)(yc * Wl + xc) * 256];
                    const float wc = (cx ? tx : 1.f - tx) * (cy ? ty : 1.f - ty);
                    s += (valid ? g : 0.f) * wc;
                }
            }
            acc += w * s;
        }
        lvl_start += Hl * Wl;
    }

    tmp[(size_t)(b * LQ + q) * 256 + h * 32 + lane] = acc;
}

// ---------------------------------------------------------------------------

extern "C" void kernel_launch(void* const* d_in, const int* in_sizes, int n_in,
                              void* d_out, int out_size, void* d_ws, size_t ws_size,
                              hipStream_t stream)
{
    const float* query = (const float*)d_in[0];
    const float* ref_l = (const float*)d_in[1];
    const float* ref_r = (const float*)d_in[2];
    const float* value = (const float*)d_in[3];
    const float* Wv    = (const float*)d_in[4];
    const float* bv    = (const float*)d_in[5];
    const float* Woff  = (const float*)d_in[6];
    const float* boff  = (const float*)d_in[7];
    const float* Watt  = (const float*)d_in[8];
    const float* batt  = (const float*)d_in[9];
    const float* Wout  = (const float*)d_in[10];
    const float* bout  = (const float*)d_in[11];
    float* out = (float*)d_out;

    const int B  = 8;
    const int LQ = 1700;
    const int L  = 22050;
    const int Mv = B * L;    // 176400
    const int Mq = B * LQ;   // 13600

    float* ws      = (float*)d_ws;
    float* v_buf   = ws;                                  // Mv*256
    float* off_buf = v_buf   + (size_t)Mv * 256;          // Mq*384
    float* att_buf = off_buf + (size_t)Mq * 384;          // Mq*96
    float* tmp_buf = att_buf + (size_t)Mq * 96;           // Mq*256

    const dim3 blk(GEMM_BLOCK);
    const int mbV = (Mv + BM - 1) / BM;                   // 2757
    const int mbQ = (Mq + BM - 1) / BM;                   // 213

    // 1) v = value @ Wv + bv        (176400 x 512 x 256)
    hipLaunchKernelGGL(wmma_gemm_f32, dim3(2, mbV), blk, 0, stream,
                       value, Wv, bv, v_buf, Mv, 256, 512);
    // 2) off = query @ Woff + boff  (13600 x 256 x 384)
    hipLaunchKernelGGL(wmma_gemm_f32, dim3(3, mbQ), blk, 0, stream,
                       query, Woff, boff, off_buf, Mq, 384, 256);
    // 3) att = query @ Watt + batt  (13600 x 256 x 96)
    hipLaunchKernelGGL(wmma_gemm_f32, dim3(1, mbQ), blk, 0, stream,
                       query, Watt, batt, att_buf, Mq, 96, 256);
    // 4) softmax over NL*NP = 12 per (b,q,h)
    {
        const int total = Mq * 8;                         // 108800
        hipLaunchKernelGGL(softmax12_kernel, dim3((total + 255) / 256), dim3(256),
                           0, stream, att_buf, total);
    }
    // 5) deformable sampling -> tmp (b*lq, 256); 1 wave per (b,h,q)
    hipLaunchKernelGGL(ms_deform_sample, dim3((B * 8 * LQ) / 8), dim3(256),
                       0, stream, v_buf, off_buf, att_buf, ref_l, ref_r, tmp_buf, LQ);
    // 6) out = tmp @ Wout + bout    (13600 x 256 x 256)
    hipLaunchKernelGGL(wmma_gemm_f32, dim3(2, mbQ), blk, 0, stream,
                       tmp_buf, Wout, bout, out, Mq, 256, 256);
}